// DirectMHPInfer_41300405518586
// MI455X (gfx1250) — compile-verified
//
#include <hip/hip_runtime.h>
#include <hip/hip_bf16.h>
#include <cstdint>

#define BATCH 32
#define NPRED 25200
#define TOPK  1000
#define CONF_THR 0.7f
#define IOU_THR  0.45f

typedef __attribute__((ext_vector_type(2))) float v2f;
typedef __attribute__((ext_vector_type(8))) float v8f;
typedef __attribute__((ext_vector_type(2))) int   v2i;
typedef __attribute__((ext_vector_type(4))) int   v4i;

#if __has_builtin(__builtin_amdgcn_global_load_async_to_lds_b64) && \
    __has_builtin(__builtin_amdgcn_global_load_async_to_lds_b128) && \
    __has_builtin(__builtin_amdgcn_s_wait_asynccnt)
#define ASYNC_LDS 1
#else
#define ASYNC_LDS 0
#endif

#define GAS __attribute__((address_space(1)))
#define LAS __attribute__((address_space(3)))

// Monotonic (order-preserving) uint mapping of IEEE float.
__device__ __forceinline__ unsigned int orderable(float f) {
    unsigned int u = __float_as_uint(f);
    return (u & 0x80000000u) ? ~u : (u | 0x80000000u);
}

__device__ __forceinline__ float score_of(const float* row) {
    float obj  = row[4];
    float conf = obj * row[5];
    bool valid = (obj > CONF_THR) && (conf > CONF_THR);
    return valid ? conf : -1.0f;
}

// ---------------------------------------------------------------------------
// Kernel 1: build sortable 64-bit keys: [ orderable(score) : (0xFFFFFFFF-n) ]
// Larger key == earlier in top_k order (desc score, asc index tie-break).
// ---------------------------------------------------------------------------
__global__ void k_keys(const float* __restrict__ pred,
                       unsigned long long* __restrict__ keys) {
    int t = blockIdx.x * blockDim.x + threadIdx.x;      // exact grid: B*N
    int n = t % NPRED;
    const float* row = pred + (size_t)t * 9;
    float s = score_of(row);
    unsigned long long key =
        ((unsigned long long)orderable(s) << 32) |
        (unsigned long long)(0xFFFFFFFFu - (unsigned int)n);
    keys[t] = key;
}

// ---------------------------------------------------------------------------
// Kernel 2: exact rank selection. rank(i) = #{j : key_j > key_i}. Keys are
// unique, so ranks are a permutation -> slots 0..TOPK-1 all written once.
// Key tiles are streamed into LDS via the CDNA5 async-to-LDS path.
// ---------------------------------------------------------------------------
#define RANK_TILE 2048
__global__ void k_rank(const float* __restrict__ pred,
                       const unsigned long long* __restrict__ keys,
                       int* __restrict__ selidx,
                       float* __restrict__ selscore) {
    __shared__ unsigned long long sk[RANK_TILE];
    const int blocksPerImg = (NPRED + 255) / 256;       // 99
    int b     = blockIdx.x / blocksPerImg;
    int chunk = blockIdx.x - b * blocksPerImg;
    int n     = chunk * 256 + threadIdx.x;
    const unsigned long long* ik = keys + (size_t)b * NPRED;
    unsigned long long mykey = (n < NPRED) ? ik[n] : 0ull;
    int rank = 0;
    for (int base = 0; base < NPRED; base += RANK_TILE) {
        int cnt = min(RANK_TILE, NPRED - base);
        __syncthreads();                                 // prior tile consumed
#if ASYNC_LDS
        for (int r = 0; r < RANK_TILE / 256; ++r) {
            int j = base + r * 256 + threadIdx.x;
            if (j < NPRED)
                __builtin_amdgcn_global_load_async_to_lds_b64(
                    (GAS v2i*)(ik + j),
                    (LAS v2i*)&sk[r * 256 + threadIdx.x], 0, 0);
        }
        __builtin_amdgcn_s_wait_asynccnt(0);
#else
        for (int r = 0; r < RANK_TILE / 256; ++r) {
            int j = base + r * 256 + threadIdx.x;
            if (j < NPRED) sk[r * 256 + threadIdx.x] = ik[j];
        }
#endif
        __syncthreads();
        #pragma unroll 8
        for (int x = 0; x < cnt; ++x)
            rank += (sk[x] > mykey) ? 1 : 0;
    }
    if (n < NPRED && rank < TOPK) {
        selidx[b * TOPK + rank]   = n;
        selscore[b * TOPK + rank] =
            score_of(pred + ((size_t)b * NPRED + n) * 9);
    }
}

// ---------------------------------------------------------------------------
// Kernel 3: gather selected rows; xywh -> xyxy as a 16x4 @ 4x4 matmul on the
// matrix pipe (V_WMMA_F32_16X16X4_F32). One wave converts 16 boxes.
// A layout (ISA 16x4 f32): lanes 0-15 hold (x,y) in v[0],v[1]; lanes 16-31
// hold (w,h). B is the constant 4x16 map; columns 0..3 = x1,y1,x2,y2.
// ---------------------------------------------------------------------------
__global__ void k_boxcvt(const float* __restrict__ pred,
                         const int* __restrict__ selidx,
                         float4* __restrict__ selbox,
                         float* __restrict__ selang) {
    __shared__ float tile[4][16][16];
    const int wavesPerImg = (TOPK + 15) / 16;            // 63
    int gwave = (blockIdx.x * blockDim.x + threadIdx.x) >> 5;
    int b  = gwave / wavesPerImg;
    int wv = gwave - b * wavesPerImg;
    int lane  = threadIdx.x & 31;
    int wslot = (threadIdx.x >> 5) & 3;
    int n  = lane & 15;
    int hi = lane >> 4;
    int k  = wv * 16 + n;
    int kk = min(k, TOPK - 1);
    int idx = selidx[b * TOPK + kk];
    const float* row = pred + ((size_t)b * NPRED + idx) * 9;

#if __has_builtin(__builtin_amdgcn_wmma_f32_16x16x4_f32)
    v2f a;
    a.x = hi ? row[2] : row[0];
    a.y = hi ? row[3] : row[1];
    v2f bm;
    if (!hi) {  // rows K=0 (x), K=1 (y) of T
        bm.x = (n == 0 || n == 2) ? 1.0f : 0.0f;
        bm.y = (n == 1 || n == 3) ? 1.0f : 0.0f;
    } else {    // rows K=2 (w), K=3 (h) of T
        bm.x = (n == 0) ? -0.5f : ((n == 2) ? 0.5f : 0.0f);
        bm.y = (n == 1) ? -0.5f : ((n == 3) ? 0.5f : 0.0f);
    }
    v8f c = {};
    v8f d = __builtin_amdgcn_wmma_f32_16x16x4_f32(
        false, a, false, bm, (short)0, c, false, false);
    #pragma unroll
    for (int r = 0; r < 8; ++r)                 // D: vgpr r, lanes>=16 -> M=r+8
        tile[wslot][r + hi * 8][n] = d[r];
#endif
    __syncthreads();
    if (hi == 0 && k < TOPK) {
#if __has_builtin(__builtin_amdgcn_wmma_f32_16x16x4_f32)
        selbox[b * TOPK + k] = make_float4(tile[wslot][n][0], tile[wslot][n][1],
                                           tile[wslot][n][2], tile[wslot][n][3]);
#else
        float x = row[0], y = row[1], w2 = row[2] * 0.5f, h2 = row[3] * 0.5f;
        selbox[b * TOPK + k] = make_float4(x - w2, y - h2, x + w2, y + h2);
#endif
        float* ang = selang + (size_t)(b * TOPK + k) * 3;
        ang[0] = row[6]; ang[1] = row[7]; ang[2] = row[8];
    }
}

// ---------------------------------------------------------------------------
// Kernel 4: IoU adjacency bitmask. Thread (i, w) builds word w of row i:
// bit jj set iff j = 32w+jj > i and IoU(i,j) > 0.45. Boxes tiled into LDS
// via async-to-LDS (16 KB per image).
// ---------------------------------------------------------------------------
__global__ void k_iou(const float4* __restrict__ selbox,
                      unsigned int* __restrict__ mask) {
    __shared__ float4 sbox[TOPK];
    const int rowBlocks = TOPK / 8;                      // 125
    int b  = blockIdx.x / rowBlocks;
    int rb = blockIdx.x - b * rowBlocks;
    const float4* ib = selbox + b * TOPK;
    int tid = threadIdx.x;
#if ASYNC_LDS
    for (int t = tid; t < TOPK; t += 256)
        __builtin_amdgcn_global_load_async_to_lds_b128(
            (GAS v4i*)(ib + t), (LAS v4i*)&sbox[t], 0, 0);
    __builtin_amdgcn_s_wait_asynccnt(0);
#else
    for (int t = tid; t < TOPK; t += 256) sbox[t] = ib[t];
#endif
    __syncthreads();
    int i = rb * 8 + (tid >> 5);
    int w = tid & 31;
    float4 bi = sbox[i];
    float areai = (bi.z - bi.x) * (bi.w - bi.y);
    unsigned int bits = 0;
    int j0 = w * 32;
    #pragma unroll
    for (int jj = 0; jj < 32; ++jj) {
        int j = j0 + jj;
        if (j < TOPK && j > i) {
            float4 bj = sbox[j];
            float areaj = (bj.z - bj.x) * (bj.w - bj.y);
            float lx = fmaxf(bi.x, bj.x), ly = fmaxf(bi.y, bj.y);
            float rx = fminf(bi.z, bj.z), ry = fminf(bi.w, bj.w);
            float iw = fmaxf(rx - lx, 0.0f), ih = fmaxf(ry - ly, 0.0f);
            float inter = iw * ih;
            float iou = inter / (areai + areaj - inter + 1e-7f);
            if (iou > IOU_THR) bits |= (1u << jj);
        }
    }
    mask[(size_t)(b * TOPK + i) * 32 + w] = bits;
}

// ---------------------------------------------------------------------------
// Kernel 5: sequential greedy suppression + output. wave32: 1000 keep bits =
// 32 words, one word per lane; per-iteration broadcast via __shfl.
// ---------------------------------------------------------------------------
__global__ void k_nms_out(const float4* __restrict__ selbox,
                          const float* __restrict__ selscore,
                          const float* __restrict__ selang,
                          const unsigned int* __restrict__ mask,
                          float* __restrict__ out,
                          float* __restrict__ keepout) {
    __shared__ float ss[TOPK];
    __shared__ unsigned int keepbits[32];
    int b = blockIdx.x;
    int tid = threadIdx.x;
    for (int k = tid; k < TOPK; k += blockDim.x) ss[k] = selscore[b * TOPK + k];
    __syncthreads();
    if (tid < 32) {                                      // one wave32
        unsigned int remv = 0, kb = 0;
        const unsigned int* mrow = mask + (size_t)b * TOPK * 32;
        for (int i = 0; i < TOPK; ++i) {
            int word = i >> 5, bit = i & 31;
            unsigned int rw = __shfl(remv, word, 32);    // uniform broadcast
            bool kp = (((rw >> bit) & 1u) == 0u) && (ss[i] > CONF_THR);
            if (kp) {                                    // uniform branch
                remv |= mrow[(size_t)i * 32 + tid];
                if (tid == word) kb |= (1u << bit);
            }
        }
        keepbits[tid] = kb;
    }
    __syncthreads();
    for (int k = tid; k < TOPK; k += blockDim.x) {
        bool kp = (keepbits[k >> 5] >> (k & 31)) & 1u;
        float4 bx = selbox[b * TOPK + k];
        const float* ang = selang + (size_t)(b * TOPK + k) * 3;
        float* o = out + (size_t)(b * TOPK + k) * 9;
        if (kp) {
            o[0] = bx.x; o[1] = bx.y; o[2] = bx.z; o[3] = bx.w;
            o[4] = ss[k]; o[5] = 0.0f;
            o[6] = ang[0]; o[7] = ang[1]; o[8] = ang[2];
        } else {
            #pragma unroll
            for (int q = 0; q < 9; ++q) o[q] = 0.0f;
        }
        keepout[b * TOPK + k] = kp ? 1.0f : 0.0f;
    }
}

// ---------------------------------------------------------------------------
extern "C" void kernel_launch(void* const* d_in, const int* in_sizes, int n_in,
                              void* d_out, int out_size, void* d_ws, size_t ws_size,
                              hipStream_t stream) {
    const float* pred = (const float*)d_in[0];
    char* ws = (char*)d_ws;
    size_t off = 0;
    unsigned long long* keys = (unsigned long long*)(ws + off);
    off += (size_t)BATCH * NPRED * 8;                    // 6,451,200 B
    int* selidx = (int*)(ws + off);        off += (size_t)BATCH * TOPK * 4;
    float* selscore = (float*)(ws + off);  off += (size_t)BATCH * TOPK * 4;
    float4* selbox = (float4*)(ws + off);  off += (size_t)BATCH * TOPK * 16;
    float* selang = (float*)(ws + off);    off += (size_t)BATCH * TOPK * 12;
    unsigned int* mask = (unsigned int*)(ws + off);
    off += (size_t)BATCH * TOPK * 32 * 4;                // total ~11.6 MB

    float* out = (float*)d_out;
    float* keepout = out + (size_t)BATCH * TOPK * 9;

    k_keys  <<<(BATCH * NPRED) / 256, 256, 0, stream>>>(pred, keys);
    k_rank  <<<BATCH * ((NPRED + 255) / 256), 256, 0, stream>>>(pred, keys, selidx, selscore);
    k_boxcvt<<<(BATCH * 63 * 32) / 128, 128, 0, stream>>>(pred, selidx, selbox, selang);
    k_iou   <<<BATCH * (TOPK / 8), 256, 0, stream>>>(selbox, mask);
    k_nms_out<<<BATCH, 1024, 0, stream>>>(selbox, selscore, selang, mask, out, keepout);
}